// AggregateUserStates_81363860455899
// MI455X (gfx1250) — compile-verified
//
#include <hip/hip_runtime.h>
#include <stdint.h>

typedef float v2f __attribute__((ext_vector_type(2)));
typedef float v4f __attribute__((ext_vector_type(4)));
typedef float v8f __attribute__((ext_vector_type(8)));
typedef int   v4i __attribute__((vector_size(16)));   // matches async builtin param type

#define BB  32
#define UU  8
#define NN  3072
#define DSZ 64
#define HH  128
#define MT  2            // M-tiles (16 tokens each) per wave -> B-fragment reuse x2
#define TOK (16 * MT)    // tokens per block

// packed-weight layout offsets (floats)
#define WP_W1 0
#define WP_W2 8192
#define WP_WO 24576
#define WP_TOTAL 32768

#if __has_builtin(__builtin_amdgcn_global_load_async_to_lds_b128) && \
    __has_builtin(__builtin_amdgcn_s_wait_asynccnt)
#define USE_ASYNC_LDS 1
typedef __attribute__((address_space(1))) v4i gas_v4i;   // global (AS1)
typedef __attribute__((address_space(3))) v4i las_v4i;   // LDS (AS3)
#else
#define USE_ASYNC_LDS 0
#endif

// ---------------------------------------------------------------------------
// One-shot weight repack into WMMA B-fragment order:
//   wp[base + ((kt*NT+nt)*32 + lane)*2 + j] = W[(kt*4 + (lane>>4)*2 + j)*NC + nt*16 + (lane&15)]
// so each lane's (wb.x, wb.y) is one aligned 8-byte load.
// ---------------------------------------------------------------------------
__global__ __launch_bounds__(256, 1)
void repack_weights_kernel(const float* __restrict__ W1,
                           const float* __restrict__ W2,
                           const float* __restrict__ Wo,
                           float* __restrict__ wp)
{
    int f = blockIdx.x * 256 + threadIdx.x;       // 0..32767
    const float* W;
    int NT, base;
    if (f < WP_W2)      { W = W1; NT = 8; base = WP_W1; }
    else if (f < WP_WO) { W = W2; NT = 8; base = WP_W2; }
    else                { W = Wo; NT = 4; base = WP_WO; }
    int local  = f - base;
    int frag   = local >> 6;
    int within = local & 63;
    int lane   = within >> 1;
    int j      = within & 1;
    int kt     = frag / NT;
    int nt     = frag % NT;
    int row    = kt * 4 + (lane >> 4) * 2 + j;
    int col    = nt * 16 + (lane & 15);
    wp[f] = W[row * (NT * 16) + col];
}

// ---------------------------------------------------------------------------
// One MLP layer on a TOKxK tile staged in LDS (row stride HH), output written
// back to the same LDS region with row stride NC = NT*16.
// ---------------------------------------------------------------------------
template<int KT, int NT, bool RELU, bool PACKED>
__device__ __forceinline__
void mlp_layer(const float* __restrict__ Wrow,   // row-major weights (fallback)
               const float* __restrict__ Wp,     // packed fragments (PACKED)
               const float* __restrict__ bias,
               float* __restrict__ actu,
               int lane, int mlo, int hi, float scale)
{
    const int NC = NT * 16;
    v8f acc[MT][NT];
    #pragma unroll
    for (int nt = 0; nt < NT; ++nt) {
        float bv = bias[nt * 16 + mlo];
        #pragma unroll
        for (int mt = 0; mt < MT; ++mt)
            acc[mt][nt] = (v8f){bv, bv, bv, bv, bv, bv, bv, bv};
    }

    const float* wpb = PACKED ? (Wp + lane * 2) : Wrow;

    #pragma unroll
    for (int kt = 0; kt < KT; ++kt) {
        const int k0 = kt * 4 + hi * 2;
        v2f a[MT];
        #pragma unroll
        for (int mt = 0; mt < MT; ++mt)
            a[mt] = *(const v2f*)(&actu[(mt * 16 + mlo) * HH + k0]);
        #pragma unroll
        for (int nt = 0; nt < NT; ++nt) {
            v2f wb;
            if (PACKED) {
                wb = *(const v2f*)(wpb + (kt * NT + nt) * 64);   // global_load_b64, imm offset
            } else {
                wb.x = Wrow[(k0    ) * NC + nt * 16 + mlo];
                wb.y = Wrow[(k0 + 1) * NC + nt * 16 + mlo];
            }
            #pragma unroll
            for (int mt = 0; mt < MT; ++mt)
                acc[mt][nt] = __builtin_amdgcn_wmma_f32_16x16x4_f32(
                    false, a[mt], false, wb, (short)0, acc[mt][nt], false, false);
        }
    }

    // epilogue: ReLU (layers 1-2) or mask-scale (layer 3); write at stride NC
    #pragma unroll
    for (int mt = 0; mt < MT; ++mt)
        #pragma unroll
        for (int nt = 0; nt < NT; ++nt)
            #pragma unroll
            for (int e = 0; e < 8; ++e) {
                int m = mt * 16 + e + 8 * hi;
                float v = acc[mt][nt][e];
                v = RELU ? fmaxf(v, 0.0f) : (v * scale);
                actu[m * NC + nt * 16 + mlo] = v;
            }
}

// ---------------------------------------------------------------------------
template<bool PACKED>
__global__ __launch_bounds__(256, 1)
void mlp_loo_agg_kernel(const float* __restrict__ s,
                        const float* __restrict__ atx,
                        const float* __restrict__ W1, const float* __restrict__ b1,
                        const float* __restrict__ W2, const float* __restrict__ b2,
                        const float* __restrict__ Wo, const float* __restrict__ bo,
                        const float* __restrict__ wp,
                        float* __restrict__ out)
{
    // Per-wave activation staging: 32 rows x 128 cols fp32 per user. 128 KB total.
    __shared__ float act[UU][TOK * HH];

    const int tid  = threadIdx.x;
    const int u    = tid >> 5;       // wave id == user index
    const int lane = tid & 31;
    const int b    = blockIdx.y;
    const int n0   = blockIdx.x * TOK;

    const int mlo = lane & 15;
    const int hi  = lane >> 4;

    float* actu = &act[u][0];

    // ---------------- stage 0: load s tile (TOK x 64) into act[u] ----------------
    {
        const float* sbase = s + (((size_t)(b * UU + u) * NN + n0) * DSZ);
        #pragma unroll
        for (int i = 0; i < (TOK * DSZ) / (32 * 4); ++i) {   // 16 iters of b128/lane
            int flat = (i * 32 + lane) * 4;                  // 0..2047
            int m = flat >> 6;
            int k = flat & 63;
#if USE_ASYNC_LDS
            __builtin_amdgcn_global_load_async_to_lds_b128(
                (gas_v4i*)(uintptr_t)(sbase + m * DSZ + k),
                (las_v4i*)(uintptr_t)(&actu[m * HH + k]),
                0, 0);
#else
            *(v4f*)(&actu[m * HH + k]) = *(const v4f*)(sbase + m * DSZ + k);
#endif
        }
#if USE_ASYNC_LDS
        __builtin_amdgcn_s_wait_asynccnt(0);
#endif
    }

    const float atx_u = atx[b * UU + u];

    mlp_layer<16, 8, true,  PACKED>(W1, PACKED ? wp + WP_W1 : wp, b1, actu, lane, mlo, hi, 1.0f);
    mlp_layer<32, 8, true,  PACKED>(W2, PACKED ? wp + WP_W2 : wp, b2, actu, lane, mlo, hi, 1.0f);
    mlp_layer<32, 4, false, PACKED>(Wo, PACKED ? wp + WP_WO : wp, bo, actu, lane, mlo, hi, atx_u);

    __syncthreads();

    // ---------------- leave-one-out reduction over U + scale ----------------
    float cnt = 0.0f;
    #pragma unroll
    for (int uu = 0; uu < UU; ++uu) cnt += atx[b * UU + uu];
    float p   = fmaxf(cnt - 1.0f, 0.0f);
    float inv = (p == 0.0f) ? 1.0f : (1.0f / p);

    #pragma unroll
    for (int h = 0; h < (TOK * DSZ) / (256 * 4); ++h) {   // 2 iters of v4 per thread
        const int base = (h * 256 + tid) * 4;             // 0..2044 over TOK x 64
        const int m = base >> 6;
        const int d = base & 63;

        v4f vals[UU];
        v4f sum = (v4f){0.0f, 0.0f, 0.0f, 0.0f};
        #pragma unroll
        for (int uu = 0; uu < UU; ++uu) {
            vals[uu] = *(const v4f*)(&act[uu][m * DSZ + d]);
            sum += vals[uu];
        }
        #pragma unroll
        for (int uu = 0; uu < UU; ++uu) {
            v4f r = (sum - vals[uu]) * inv;
            *(v4f*)(out + (((size_t)(b * UU + uu) * NN + (n0 + m)) * DSZ + d)) = r;
        }
    }
}

extern "C" void kernel_launch(void* const* d_in, const int* in_sizes, int n_in,
                              void* d_out, int out_size, void* d_ws, size_t ws_size,
                              hipStream_t stream) {
    const float* s   = (const float*)d_in[0];
    const float* atx = (const float*)d_in[1];
    const float* W1  = (const float*)d_in[2];
    const float* b1  = (const float*)d_in[3];
    const float* W2  = (const float*)d_in[4];
    const float* b2  = (const float*)d_in[5];
    const float* Wo  = (const float*)d_in[6];
    const float* bo  = (const float*)d_in[7];
    float* out = (float*)d_out;

    dim3 grid(NN / TOK, BB, 1);

    if (ws_size >= (size_t)WP_TOTAL * sizeof(float)) {
        float* wp = (float*)d_ws;
        repack_weights_kernel<<<dim3(WP_TOTAL / 256, 1, 1), dim3(256, 1, 1), 0, stream>>>(
            W1, W2, Wo, wp);
        mlp_loo_agg_kernel<true><<<grid, dim3(256, 1, 1), 0, stream>>>(
            s, atx, W1, b1, W2, b2, Wo, bo, wp, out);
    } else {
        mlp_loo_agg_kernel<false><<<grid, dim3(256, 1, 1), 0, stream>>>(
            s, atx, W1, b1, W2, b2, Wo, bo, nullptr, out);
    }
}